// HgtConv_54528904790449
// MI455X (gfx1250) — compile-verified
//
#include <hip/hip_runtime.h>
#include <math.h>

#define N_NODES 50000
#define N_EDGES 300000
#define CDIM    256
#define HEADS   8
#define DDIM    32
#define NTYPES  3
#define MTILE   32        // rows per workgroup (2 accumulators/wave -> B reuse x2)

typedef float v2f __attribute__((ext_vector_type(2)));
typedef float v8f __attribute__((ext_vector_type(8)));

#define LDSA_STRIDE 260   // 260 % 64 == 4 -> rows hit distinct banks

// order-preserving float <-> uint transform for atomicMax-based segment max
__device__ __forceinline__ unsigned flip_f(float f) {
  unsigned u = __float_as_uint(f);
  return (u & 0x80000000u) ? ~u : (u | 0x80000000u);
}
__device__ __forceinline__ float unflip_f(unsigned u) {
  unsigned b = (u & 0x80000000u) ? (u & 0x7FFFFFFFu) : ~u;
  return __uint_as_float(b);
}

// ---------------------------------------------------------------------------
// Type-bucketing: counting sort of node ids by ntype (3 buckets) so GEMM tiles
// are type-uniform -> single WMMA pass, no mask VALU.
// ---------------------------------------------------------------------------
__global__ __launch_bounds__(256)
void hgt_count_types(const int* __restrict__ ntype, int* __restrict__ cnt)
{
  const int i = blockIdx.x * 256 + threadIdx.x;
  if (i < N_NODES) atomicAdd(&cnt[ntype[i]], 1);
}

__global__ void hgt_scan3(const int* __restrict__ cnt, int* __restrict__ cursor)
{
  cursor[0] = 0;
  cursor[1] = cnt[0];
  cursor[2] = cnt[0] + cnt[1];
}

__global__ __launch_bounds__(256)
void hgt_scatter_perm(const int* __restrict__ ntype, int* __restrict__ cursor,
                      int* __restrict__ perm)
{
  const int i = blockIdx.x * 256 + threadIdx.x;
  if (i >= N_NODES) return;
  const int t = ntype[i];
  const int p = atomicAdd(&cursor[t], 1);
  perm[p] = i;
}

// ---------------------------------------------------------------------------
// Kernel 1: typed linear k/q/v = x @ W[ntype] via V_WMMA_F32_16X16X4_F32 over
// type-sorted rows. Workgroup = 32 permuted rows (4 waves x 4 col-tiles of 16,
// 2 accumulators per wave so each B fragment feeds 2 WMMAs).
// ---------------------------------------------------------------------------
__global__ __launch_bounds__(128)
void hgt_typed_qkv(const float* __restrict__ x,
                   const float* __restrict__ Wk,
                   const float* __restrict__ Wq,
                   const float* __restrict__ Wv,
                   const int*   __restrict__ ntype,
                   const int*   __restrict__ perm,
                   float* __restrict__ Kout,
                   float* __restrict__ Qout,
                   float* __restrict__ Vout)
{
  __shared__ float sA[MTILE * LDSA_STRIDE];
  __shared__ int   sT[MTILE];
  __shared__ int   sP[MTILE];

  const int rowBase = blockIdx.x * MTILE;
  const float* W;
  float* Out;
  if (blockIdx.z == 0)      { W = Wk; Out = Kout; }
  else if (blockIdx.z == 1) { W = Wq; Out = Qout; }
  else                      { W = Wv; Out = Vout; }

  const int tid = threadIdx.x;
  if (tid < MTILE) {
    int idx = rowBase + tid;
    if (idx > N_NODES - 1) idx = N_NODES - 1;   // clamp tail (duplicates own rows)
    const int p = perm[idx];
    sP[tid] = p;
    sT[tid] = ntype[p];
  }
  __syncthreads();

  // cooperative float4 fill of the 32x256 A tile (gather through perm)
  for (int q = tid; q < MTILE * 64; q += 128) {
    const int r  = q >> 6;
    const int c4 = q & 63;
    float4 val = ((const float4*)(x + (size_t)sP[r] * CDIM))[c4];
    *((float4*)(sA + r * LDSA_STRIDE + c4 * 4)) = val;
  }
  __syncthreads();

  const int t0 = sT[0];
  bool uni = true;
  #pragma unroll
  for (int r = 1; r < MTILE; ++r) uni = uni && (sT[r] == t0);

  const int lane = tid & 31;
  const int wave = tid >> 5;
  const int half = lane >> 4;   // lane 0-15 vs 16-31
  const int l16  = lane & 15;

  for (int i = 0; i < 4; ++i) {
    const int ct   = wave * 4 + i;
    const int colg = ct * 16 + l16;
    v8f c0 = {0.f, 0.f, 0.f, 0.f, 0.f, 0.f, 0.f, 0.f};
    v8f c1 = {0.f, 0.f, 0.f, 0.f, 0.f, 0.f, 0.f, 0.f};

    if (uni) {
      const float* Wt = W + (size_t)t0 * CDIM * CDIM;
      __builtin_prefetch(Wt + (size_t)0 * CDIM + colg, 0, 1);
      __builtin_prefetch(Wt + (size_t)128 * CDIM + colg, 0, 1);
      for (int k0 = 0; k0 < CDIM; k0 += 4) {
        const int ka = k0 + half * 2;
        v2f b, a0, a1;
        b.x  = Wt[(size_t)ka       * CDIM + colg];
        b.y  = Wt[(size_t)(ka + 1) * CDIM + colg];
        a0.x = sA[l16 * LDSA_STRIDE + ka];
        a0.y = sA[l16 * LDSA_STRIDE + ka + 1];
        a1.x = sA[(16 + l16) * LDSA_STRIDE + ka];
        a1.y = sA[(16 + l16) * LDSA_STRIDE + ka + 1];
        c0 = __builtin_amdgcn_wmma_f32_16x16x4_f32(false, a0, false, b,
                                                   (short)0, c0, false, false);
        c1 = __builtin_amdgcn_wmma_f32_16x16x4_f32(false, a1, false, b,
                                                   (short)0, c1, false, false);
      }
    } else {  // rare type-boundary tile: masked 3-pass accumulation
      for (int t = 0; t < NTYPES; ++t) {
        const float m0 = (sT[l16] == t)      ? 1.0f : 0.0f;
        const float m1 = (sT[16 + l16] == t) ? 1.0f : 0.0f;
        const float* Wt = W + (size_t)t * CDIM * CDIM;
        for (int k0 = 0; k0 < CDIM; k0 += 4) {
          const int ka = k0 + half * 2;
          v2f b, a0, a1;
          b.x  = Wt[(size_t)ka       * CDIM + colg];
          b.y  = Wt[(size_t)(ka + 1) * CDIM + colg];
          a0.x = sA[l16 * LDSA_STRIDE + ka]            * m0;
          a0.y = sA[l16 * LDSA_STRIDE + ka + 1]        * m0;
          a1.x = sA[(16 + l16) * LDSA_STRIDE + ka]     * m1;
          a1.y = sA[(16 + l16) * LDSA_STRIDE + ka + 1] * m1;
          c0 = __builtin_amdgcn_wmma_f32_16x16x4_f32(false, a0, false, b,
                                                     (short)0, c0, false, false);
          c1 = __builtin_amdgcn_wmma_f32_16x16x4_f32(false, a1, false, b,
                                                     (short)0, c1, false, false);
        }
      }
    }
    #pragma unroll
    for (int j = 0; j < 8; ++j) {
      const int r0 = j + half * 8;          // local rows 0..15 (always valid)
      const int r1 = 16 + j + half * 8;     // local rows 16..31 (tail-guarded)
      Out[(size_t)sP[r0] * CDIM + colg] = c0[j];
      if (rowBase + r1 < N_NODES)
        Out[(size_t)sP[r1] * CDIM + colg] = c1[j];
    }
  }
}

// ---------------------------------------------------------------------------
// Kernel 2: per-(edge,head) attention logit a = k^T * Ratt[r,h] * q * pri / sqrt(D)
// and atomic segment-max into amaxU (flipped-uint). K/Q are L2-resident (154MB<192MB).
// ---------------------------------------------------------------------------
__global__ __launch_bounds__(256)
void hgt_edge_logits(const float* __restrict__ Kf, const float* __restrict__ Qf,
                     const float* __restrict__ Ratt, const float* __restrict__ pri,
                     const int* __restrict__ src, const int* __restrict__ dst,
                     const int* __restrict__ etype,
                     float* __restrict__ aE, unsigned* __restrict__ amaxU)
{
  const int tid = blockIdx.x * 256 + threadIdx.x;
  if (tid >= N_EDGES * HEADS) return;
  const int e = tid >> 3, h = tid & 7;
  const int s = src[e], d = dst[e], r = etype[e];

  const float* kv = Kf + (size_t)s * CDIM + h * DDIM;
  const float* qv = Qf + (size_t)d * CDIM + h * DDIM;
  const float* R  = Ratt + (size_t)(r * HEADS + h) * DDIM * DDIM;

  float qreg[DDIM];
  #pragma unroll
  for (int f = 0; f < DDIM; ++f) qreg[f] = qv[f];

  float acc = 0.f;                       // a = sum_d k_d * (R q)_d
  for (int dd = 0; dd < DDIM; ++dd) {
    const float* Rr = R + dd * DDIM;
    float tmp = 0.f;
    #pragma unroll
    for (int f = 0; f < DDIM; ++f) tmp = fmaf(Rr[f], qreg[f], tmp);
    acc = fmaf(kv[dd], tmp, acc);
  }
  const float a = acc * pri[r * HEADS + h] * 0.17677669529663688f; // 1/sqrt(32)
  aE[tid] = a;
  atomicMax(&amaxU[d * HEADS + h], flip_f(a));
}

// ---------------------------------------------------------------------------
// Kernel 3: softmax denominators per (dst,head)
// ---------------------------------------------------------------------------
__global__ __launch_bounds__(256)
void hgt_edge_denom(const float* __restrict__ aE,
                    const unsigned* __restrict__ amaxU,
                    const int* __restrict__ dst,
                    float* __restrict__ denom)
{
  const int tid = blockIdx.x * 256 + threadIdx.x;
  if (tid >= N_EDGES * HEADS) return;
  const int e = tid >> 3, h = tid & 7;
  const int d = dst[e];
  const float am = unflip_f(amaxU[d * HEADS + h]);
  const float ex = __expf(aE[tid] - am);
  atomicAdd(&denom[d * HEADS + h], ex);
}

// ---------------------------------------------------------------------------
// Kernel 4: mv = v^T * Rmsg[r,h]; hagg[dst] += attn * mv (atomic scatter-sum)
// ---------------------------------------------------------------------------
__global__ __launch_bounds__(256)
void hgt_edge_aggregate(const float* __restrict__ Vf, const float* __restrict__ Rmsg,
                        const float* __restrict__ aE,
                        const unsigned* __restrict__ amaxU,
                        const float* __restrict__ denom,
                        const int* __restrict__ src, const int* __restrict__ dst,
                        const int* __restrict__ etype,
                        float* __restrict__ hagg)
{
  const int tid = blockIdx.x * 256 + threadIdx.x;
  if (tid >= N_EDGES * HEADS) return;
  const int e = tid >> 3, h = tid & 7;
  const int s = src[e], d = dst[e], r = etype[e];

  const float am   = unflip_f(amaxU[d * HEADS + h]);
  const float attn = __expf(aE[tid] - am) / denom[d * HEADS + h];

  const float* vv = Vf + (size_t)s * CDIM + h * DDIM;
  const float* R  = Rmsg + (size_t)(r * HEADS + h) * DDIM * DDIM;

  float mv[DDIM];
  #pragma unroll
  for (int f = 0; f < DDIM; ++f) mv[f] = 0.f;
  for (int dd = 0; dd < DDIM; ++dd) {
    const float vd = vv[dd];
    const float* Rr = R + dd * DDIM;
    #pragma unroll
    for (int f = 0; f < DDIM; ++f) mv[f] = fmaf(vd, Rr[f], mv[f]);
  }
  float* outp = hagg + (size_t)d * CDIM + h * DDIM;
  #pragma unroll
  for (int f = 0; f < DDIM; ++f) atomicAdd(&outp[f], attn * mv[f]);
}

// ---------------------------------------------------------------------------
// Kernel 5: out = sigmoid(skip[t]) * (h @ Wa[t]) + (1-alpha) * x, in-place on
// d_out, over type-sorted rows. Workgroup owns its 32 permuted rows entirely
// (tail clamp duplicates only the tile's own rows) -> in-place is race-free.
// ---------------------------------------------------------------------------
__global__ __launch_bounds__(128)
void hgt_out_proj(float* __restrict__ hout,           // in: aggregate, out: final
                  const float* __restrict__ x,
                  const float* __restrict__ Wa,
                  const float* __restrict__ skip,
                  const int*   __restrict__ ntype,
                  const int*   __restrict__ perm)
{
  __shared__ float sA[MTILE * LDSA_STRIDE];
  __shared__ int   sT[MTILE];
  __shared__ int   sP[MTILE];
  __shared__ float sAl[MTILE];

  const int rowBase = blockIdx.x * MTILE;
  const int tid = threadIdx.x;

  if (tid < MTILE) {
    int idx = rowBase + tid;
    if (idx > N_NODES - 1) idx = N_NODES - 1;
    const int p = perm[idx];
    const int t = ntype[p];
    sP[tid]  = p;
    sT[tid]  = t;
    sAl[tid] = 1.0f / (1.0f + __expf(-skip[t]));
  }
  __syncthreads();

  for (int q = tid; q < MTILE * 64; q += 128) {
    const int r  = q >> 6;
    const int c4 = q & 63;
    float4 val = ((const float4*)(hout + (size_t)sP[r] * CDIM))[c4];
    *((float4*)(sA + r * LDSA_STRIDE + c4 * 4)) = val;
  }
  __syncthreads();

  const int t0 = sT[0];
  bool uni = true;
  #pragma unroll
  for (int r = 1; r < MTILE; ++r) uni = uni && (sT[r] == t0);

  const int lane = tid & 31;
  const int wave = tid >> 5;
  const int half = lane >> 4;
  const int l16  = lane & 15;

  for (int i = 0; i < 4; ++i) {
    const int ct   = wave * 4 + i;
    const int colg = ct * 16 + l16;
    v8f c0 = {0.f, 0.f, 0.f, 0.f, 0.f, 0.f, 0.f, 0.f};
    v8f c1 = {0.f, 0.f, 0.f, 0.f, 0.f, 0.f, 0.f, 0.f};

    if (uni) {
      const float* Wt = Wa + (size_t)t0 * CDIM * CDIM;
      __builtin_prefetch(Wt + (size_t)0 * CDIM + colg, 0, 1);
      __builtin_prefetch(Wt + (size_t)128 * CDIM + colg, 0, 1);
      for (int k0 = 0; k0 < CDIM; k0 += 4) {
        const int ka = k0 + half * 2;
        v2f b, a0, a1;
        b.x  = Wt[(size_t)ka       * CDIM + colg];
        b.y  = Wt[(size_t)(ka + 1) * CDIM + colg];
        a0.x = sA[l16 * LDSA_STRIDE + ka];
        a0.y = sA[l16 * LDSA_STRIDE + ka + 1];
        a1.x = sA[(16 + l16) * LDSA_STRIDE + ka];
        a1.y = sA[(16 + l16) * LDSA_STRIDE + ka + 1];
        c0 = __builtin_amdgcn_wmma_f32_16x16x4_f32(false, a0, false, b,
                                                   (short)0, c0, false, false);
        c1 = __builtin_amdgcn_wmma_f32_16x16x4_f32(false, a1, false, b,
                                                   (short)0, c1, false, false);
      }
    } else {
      for (int t = 0; t < NTYPES; ++t) {
        const float m0 = (sT[l16] == t)      ? 1.0f : 0.0f;
        const float m1 = (sT[16 + l16] == t) ? 1.0f : 0.0f;
        const float* Wt = Wa + (size_t)t * CDIM * CDIM;
        for (int k0 = 0; k0 < CDIM; k0 += 4) {
          const int ka = k0 + half * 2;
          v2f b, a0, a1;
          b.x  = Wt[(size_t)ka       * CDIM + colg];
          b.y  = Wt[(size_t)(ka + 1) * CDIM + colg];
          a0.x = sA[l16 * LDSA_STRIDE + ka]            * m0;
          a0.y = sA[l16 * LDSA_STRIDE + ka + 1]        * m0;
          a1.x = sA[(16 + l16) * LDSA_STRIDE + ka]     * m1;
          a1.y = sA[(16 + l16) * LDSA_STRIDE + ka + 1] * m1;
          c0 = __builtin_amdgcn_wmma_f32_16x16x4_f32(false, a0, false, b,
                                                     (short)0, c0, false, false);
          c1 = __builtin_amdgcn_wmma_f32_16x16x4_f32(false, a1, false, b,
                                                     (short)0, c1, false, false);
        }
      }
    }
    #pragma unroll
    for (int j = 0; j < 8; ++j) {
      const int r0 = j + half * 8;
      const int r1 = 16 + j + half * 8;
      {
        const int row = sP[r0];
        const float al = sAl[r0];
        const float xv = x[(size_t)row * CDIM + colg];
        hout[(size_t)row * CDIM + colg] = c0[j] * al + xv * (1.0f - al);
      }
      if (rowBase + r1 < N_NODES) {
        const int row = sP[r1];
        const float al = sAl[r1];
        const float xv = x[(size_t)row * CDIM + colg];
        hout[(size_t)row * CDIM + colg] = c1[j] * al + xv * (1.0f - al);
      }
    }
  }
}

// ---------------------------------------------------------------------------
extern "C" void kernel_launch(void* const* d_in, const int* in_sizes, int n_in,
                              void* d_out, int out_size, void* d_ws, size_t ws_size,
                              hipStream_t stream)
{
  const float* x    = (const float*)d_in[0];
  const float* Wk   = (const float*)d_in[1];
  const float* Wq   = (const float*)d_in[2];
  const float* Wv   = (const float*)d_in[3];
  const float* Ratt = (const float*)d_in[4];
  const float* Rmsg = (const float*)d_in[5];
  const float* pri  = (const float*)d_in[6];
  const float* Wa   = (const float*)d_in[7];
  const float* skip = (const float*)d_in[8];
  const int*   src  = (const int*)d_in[9];
  const int*   dst  = (const int*)d_in[10];
  const int*   etyp = (const int*)d_in[11];
  const int*   ntyp = (const int*)d_in[12];

  // workspace layout (f32): K | Q | V | edge logits | amax | denom | perm | cnt | cursor
  float*    Kf    = (float*)d_ws;
  float*    Qf    = Kf + (size_t)N_NODES * CDIM;
  float*    Vf    = Qf + (size_t)N_NODES * CDIM;
  float*    aE    = Vf + (size_t)N_NODES * CDIM;
  unsigned* amaxU = (unsigned*)(aE + (size_t)N_EDGES * HEADS);
  float*    denom = (float*)(amaxU + (size_t)N_NODES * HEADS);
  int*      perm  = (int*)(denom + (size_t)N_NODES * HEADS);
  int*      cnt   = perm + N_NODES;     // 3 ints
  int*      curs  = cnt + NTYPES;       // 3 ints

  float* hagg = (float*)d_out;  // aggregate accumulates here, projected in-place

  hipMemsetAsync(amaxU, 0, (size_t)N_NODES * HEADS * sizeof(unsigned), stream);
  hipMemsetAsync(denom, 0, (size_t)N_NODES * HEADS * sizeof(float), stream);
  hipMemsetAsync(cnt,   0, NTYPES * sizeof(int), stream);
  hipMemsetAsync(d_out, 0, (size_t)N_NODES * CDIM * sizeof(float), stream);

  // type-bucketing permutation (counting sort, 3 buckets)
  const int gN = (N_NODES + 255) / 256;
  hgt_count_types<<<gN, 256, 0, stream>>>(ntyp, cnt);
  hgt_scan3<<<1, 1, 0, stream>>>(cnt, curs);
  hgt_scatter_perm<<<gN, 256, 0, stream>>>(ntyp, curs, perm);

  const int gM = (N_NODES + MTILE - 1) / MTILE;   // 1563 (last tile half-height)
  dim3 gQKV(gM, 1, 3);
  hgt_typed_qkv<<<gQKV, 128, 0, stream>>>(x, Wk, Wq, Wv, ntyp, perm, Kf, Qf, Vf);

  const int nEH = N_EDGES * HEADS;
  const int gE  = (nEH + 255) / 256;
  hgt_edge_logits<<<gE, 256, 0, stream>>>(Kf, Qf, Ratt, pri, src, dst, etyp, aE, amaxU);
  hgt_edge_denom<<<gE, 256, 0, stream>>>(aE, amaxU, dst, denom);
  hgt_edge_aggregate<<<gE, 256, 0, stream>>>(Vf, Rmsg, aE, amaxU, denom,
                                             src, dst, etyp, hagg);

  hgt_out_proj<<<gM, 128, 0, stream>>>(hagg, x, Wa, skip, ntyp, perm);
}